// DynamicalF_CrossAttention_6485400617069
// MI455X (gfx1250) — compile-verified
//
#include <hip/hip_runtime.h>
#include <hip/hip_bf16.h>

// CDNA5 / gfx1250, wave32.
typedef float v2f __attribute__((ext_vector_type(2)));
typedef float v8f __attribute__((ext_vector_type(8)));

__device__ inline float wred32(float v) {
#pragma unroll
    for (int m = 16; m > 0; m >>= 1) v += __shfl_xor(v, m, 32);
    return v;  // full-wave sum broadcast to all lanes
}

// ---------------------------------------------------------------------------
// Precompute kernel (1 wave). Folds all weight tensors into 11 scalars using
// V_WMMA_F32_16X16X4_F32 for the real GEMM that remains:
//   Y = [Wi_AB ; Wi_BA] (192x32)  @  X (32x4),  X = [wA | bA | wB | bB]
// A-layout (ISA 7.12.2, 32-bit A 16x4): lanes = M, vgpr0/1 = K{0,1} (lanes<16)
// or K{2,3} (lanes>=16). B-layout (4x16): lanes%16 = N, same K mapping.
// ---------------------------------------------------------------------------
__global__ __launch_bounds__(32) void precompute_coeffs(
    const float* __restrict__ WA,    const float* __restrict__ bA,
    const float* __restrict__ WB,    const float* __restrict__ bB,
    const float* __restrict__ Wi_AB, const float* __restrict__ bi_AB,
    const float* __restrict__ Wo_AB, const float* __restrict__ bo_AB,
    const float* __restrict__ Wi_BA, const float* __restrict__ bi_BA,
    const float* __restrict__ Wo_BA, const float* __restrict__ bo_BA,
    const float* __restrict__ Wf,    const float* __restrict__ bf,
    float* __restrict__ coef)
{
    __shared__ float Y[192 * 16];
    const int lane  = threadIdx.x;        // 0..31, single wave
    const int n     = lane & 15;          // N column / M row within tile
    const int khalf = (lane >> 4) << 1;   // 0 or 2

    // B operand tiles for all 8 K-steps: column n holds X[:,n] (zero for n>=4)
    float Bx[8], By[8];
#pragma unroll
    for (int k0 = 0; k0 < 8; ++k0) {
        const int kb = 4 * k0 + khalf;
        float x0 = 0.f, x1 = 0.f;
        if      (n == 0) { x0 = WA[kb]; x1 = WA[kb + 1]; }
        else if (n == 1) { x0 = bA[kb]; x1 = bA[kb + 1]; }
        else if (n == 2) { x0 = WB[kb]; x1 = WB[kb + 1]; }
        else if (n == 3) { x0 = bB[kb]; x1 = bB[kb + 1]; }
        Bx[k0] = x0; By[k0] = x1;
    }

    for (int t = 0; t < 12; ++t) {   // 12 M-tiles of 16 rows (AB then BA)
        const float* Wrow = (t < 6) ? (Wi_AB + (16 * t + n) * 32)
                                    : (Wi_BA + (16 * (t - 6) + n) * 32);
        v8f acc = {0.f, 0.f, 0.f, 0.f, 0.f, 0.f, 0.f, 0.f};
#pragma unroll
        for (int k0 = 0; k0 < 8; ++k0) {
            const int kb = 4 * k0 + khalf;
            v2f a; a.x = Wrow[kb]; a.y = Wrow[kb + 1];
            v2f b; b.x = Bx[k0];   b.y = By[k0];
            acc = __builtin_amdgcn_wmma_f32_16x16x4_f32(
                false, a, false, b, (short)0, acc, false, false);
        }
        // D layout: vgpr j, lanes<16 -> M=j, lanes>=16 -> M=j+8; N=lane%16
#pragma unroll
        for (int j = 0; j < 8; ++j) {
            const int M = 16 * t + j + ((lane >> 4) << 3);
            Y[M * 16 + n] = acc[j];
        }
    }
    __syncthreads();

    const int i = lane;
    // crossAB: query = A tokens (wA/bA -> cols 0/1), kv = B tokens (cols 2/3)
    float uq  = Y[(0   + i) * 16 + 0];
    float cq  = Y[(0   + i) * 16 + 1] + bi_AB[i];
    float uk  = Y[(32  + i) * 16 + 2];
    float ck  = Y[(32  + i) * 16 + 3] + bi_AB[32 + i];
    float uv  = Y[(64  + i) * 16 + 2];
    float cv  = Y[(64  + i) * 16 + 3] + bi_AB[64 + i];
    // crossBA: query = B tokens (cols 2/3), kv = A tokens (cols 0/1)
    float uq2 = Y[(96  + i) * 16 + 2];
    float cq2 = Y[(96  + i) * 16 + 3] + bi_BA[i];
    float uk2 = Y[(128 + i) * 16 + 0];
    float ck2 = Y[(128 + i) * 16 + 1] + bi_BA[32 + i];
    float uv2 = Y[(160 + i) * 16 + 0];
    float cv2 = Y[(160 + i) * 16 + 1] + bi_BA[64 + i];

    // g = Wf @ Wo  (folds out_proj + final linear), lane i owns column i
    float gAi = 0.f, gBi = 0.f;
#pragma unroll 4
    for (int mm = 0; mm < 32; ++mm) {
        const float w = Wf[mm];
        gAi += w * Wo_AB[mm * 32 + i];
        gBi += w * Wo_BA[mm * 32 + i];
    }

    const float aAB = wred32(uq  * uk );
    const float bAB = wred32(uq  * ck );
    const float gAB = wred32(cq  * uk );
    const float dAB = wred32(cq  * ck );
    const float aBA = wred32(uq2 * uk2);
    const float bBA = wred32(uq2 * ck2);
    const float gBA = wred32(cq2 * uk2);
    const float dBA = wred32(cq2 * ck2);
    const float sA  = wred32(gAi * uv );
    const float sB  = wred32(gBi * uv2);
    const float tA  = wred32(gAi * cv ) + wred32(Wf[i] * bo_AB[i]);
    const float tB  = wred32(gBi * cv2) + wred32(Wf[i] * bo_BA[i]);

    if (lane == 0) {
        coef[0] = aAB; coef[1] = bAB; coef[2] = gAB; coef[3] = dAB;
        coef[4] = aBA; coef[5] = bBA; coef[6] = gBA; coef[7] = dBA;
        coef[8] = sA;  coef[9] = sB;
        coef[10] = 0.5f * (tA + tB) + bf[0];
    }
}

// ---------------------------------------------------------------------------
// Main streaming kernel: 20 B in + 4 B out per batch, memory-bound.
// ---------------------------------------------------------------------------
__device__ inline float softmax3dot(float c, float a0, float a1, float a2) {
    const float e0 = c * a0, e1 = c * a1, e2 = c * a2;
    const float mx = fmaxf(e0, fmaxf(e1, e2));
    const float w0 = __expf(e0 - mx), w1 = __expf(e1 - mx), w2 = __expf(e2 - mx);
    return (w0 * a0 + w1 * a1 + w2 * a2) / (w0 + w1 + w2);
}

__device__ inline float one_batch(float a0, float a1, float a2,
                                  float b0, float b1,
                                  float aAB, float gAB, float aBA, float gBA,
                                  float sA, float sB, float c0)
{
    const float inv = 0.17677669529663687f;  // 1/sqrt(32)
    // crossAB: softmax over 2 kv tokens -> sigmoid of score difference
    const float db = inv * (b1 - b0);
    const float z0 = db * (aAB * a0 + gAB);
    const float z1 = db * (aAB * a1 + gAB);
    const float z2 = db * (aAB * a2 + gAB);
    const float S  = 1.f / (1.f + __expf(-z0))
                   + 1.f / (1.f + __expf(-z1))
                   + 1.f / (1.f + __expf(-z2));
    const float MA = b0 + (b1 - b0) * (S * (1.f / 3.f));
    // crossBA: softmax over 3 kv tokens
    const float c0t = inv * (aBA * b0 + gBA);
    const float c1t = inv * (aBA * b1 + gBA);
    const float MB  = 0.5f * (softmax3dot(c0t, a0, a1, a2) +
                              softmax3dot(c1t, a0, a1, a2));
    const float o = 0.5f * (MA * sA + MB * sB) + c0;
    return (o >= 0.f) ? o : 0.01f * o;  // leaky relu
}

__global__ __launch_bounds__(256) void dynf_main(
    const float* __restrict__ gA, const float* __restrict__ gB,
    const float* __restrict__ coef, float* __restrict__ out, int nB)
{
    const float aAB = coef[0], gAB = coef[2];
    const float aBA = coef[4], gBA = coef[6];
    const float sA  = coef[8], sB  = coef[9], c0 = coef[10];

    const int idx = blockIdx.x * blockDim.x + threadIdx.x;
    const int b0  = idx * 4;  // 4 batches per thread -> aligned float4 traffic
    if (b0 + 3 < nB) {
        const float4 va0 = ((const float4*)(gA + 3 * b0))[0];
        const float4 va1 = ((const float4*)(gA + 3 * b0))[1];
        const float4 va2 = ((const float4*)(gA + 3 * b0))[2];
        const float4 vb0 = ((const float4*)(gB + 2 * b0))[0];
        const float4 vb1 = ((const float4*)(gB + 2 * b0))[1];
        float4 r;
        r.x = one_batch(va0.x, va0.y, va0.z, vb0.x, vb0.y, aAB, gAB, aBA, gBA, sA, sB, c0);
        r.y = one_batch(va0.w, va1.x, va1.y, vb0.z, vb0.w, aAB, gAB, aBA, gBA, sA, sB, c0);
        r.z = one_batch(va1.z, va1.w, va2.x, vb1.x, vb1.y, aAB, gAB, aBA, gBA, sA, sB, c0);
        r.w = one_batch(va2.y, va2.z, va2.w, vb1.z, vb1.w, aAB, gAB, aBA, gBA, sA, sB, c0);
        ((float4*)(out + b0))[0] = r;
    } else {
        for (int b = b0; b < nB; ++b) {
            out[b] = one_batch(gA[3 * b], gA[3 * b + 1], gA[3 * b + 2],
                               gB[2 * b], gB[2 * b + 1],
                               aAB, gAB, aBA, gBA, sA, sB, c0);
        }
    }
}

extern "C" void kernel_launch(void* const* d_in, const int* in_sizes, int n_in,
                              void* d_out, int out_size, void* d_ws, size_t ws_size,
                              hipStream_t stream) {
    const float* groupA = (const float*)d_in[0];
    const float* groupB = (const float*)d_in[1];
    const float* WA     = (const float*)d_in[2];
    const float* bA     = (const float*)d_in[3];
    const float* WB     = (const float*)d_in[4];
    const float* bB     = (const float*)d_in[5];
    const float* Wi_AB  = (const float*)d_in[6];
    const float* bi_AB  = (const float*)d_in[7];
    const float* Wo_AB  = (const float*)d_in[8];
    const float* bo_AB  = (const float*)d_in[9];
    const float* Wi_BA  = (const float*)d_in[10];
    const float* bi_BA  = (const float*)d_in[11];
    const float* Wo_BA  = (const float*)d_in[12];
    const float* bo_BA  = (const float*)d_in[13];
    const float* Wf     = (const float*)d_in[14];
    const float* bf     = (const float*)d_in[15];
    float* coef = (float*)d_ws;
    float* out  = (float*)d_out;

    precompute_coeffs<<<1, 32, 0, stream>>>(WA, bA, WB, bB,
                                            Wi_AB, bi_AB, Wo_AB, bo_AB,
                                            Wi_BA, bi_BA, Wo_BA, bo_BA,
                                            Wf, bf, coef);

    const int nB       = in_sizes[0] / 3;          // 524288
    const int nThreads = (nB + 3) / 4;
    const int nBlocks  = (nThreads + 255) / 256;
    dynf_main<<<nBlocks, 256, 0, stream>>>(groupA, groupB, coef, out, nB);
}